// ZBLBasis_15968688406954
// MI455X (gfx1250) — compile-verified
//
#include <hip/hip_runtime.h>
#include <hip/hip_bf16.h>

// ---------------------------------------------------------------------------
// ZBL pair potential, MI455X (gfx1250).
// Memory/atomic-bound scatter-reduce; HBM floor ~3.3us (76.8MB streams at
// 23.3 TB/s). WMMA does not apply (no dense MACC structure). CDNA5 paths:
//   - GLOBAL_LOAD_ASYNC_TO_LDS_B128 (ASYNCcnt) double-buffered stream staging
//   - s_wait_asynccnt pipelining
//   - L2-resident cached gathers of the per-node table
//   - global_atomic_add_f32 (no-return, STOREcnt), skipped past the cutoff
// ---------------------------------------------------------------------------

#define KE_CONST 14.3996454784255f
#define INV_A_PREF (1.0f / (0.4543f * 0.529f))

#define BLOCK_T 256
#define EDGES_PER_T 4
#define TILE_EDGES (BLOCK_T * EDGES_PER_T)   // 1024 edges / tile

// -------------------------- per-node precompute ----------------------------
__global__ void zbl_node_prep(const float* __restrict__ node_attrs,
                              const int*   __restrict__ atomic_numbers,
                              const float* __restrict__ covalent_radii,
                              float4*      __restrict__ nodeinfo,
                              float*       __restrict__ out,
                              int n_nodes, int n_elem) {
    int i = blockIdx.x * blockDim.x + threadIdx.x;
    if (i >= n_nodes) return;

    const float* row = node_attrs + (size_t)i * (size_t)n_elem;
    float best = row[0];
    int   arg  = 0;
    for (int j = 1; j < n_elem; ++j) {
        float v = row[j];
        if (v > best) { best = v; arg = j; }   // first-max, matches jnp.argmax
    }
    int   Z    = atomic_numbers[arg];
    float Zf   = (float)Z;
    float Zpow = powf(Zf, 0.3f);
    float rad  = covalent_radii[Z];

    nodeinfo[i] = make_float4(Zf, Zpow, rad, 0.0f);
    out[i] = 0.0f;                              // d_out is poisoned by harness
}

// ----------------------------- edge compute --------------------------------
__device__ __forceinline__ void zbl_edge_compute(float xv, int s, int r,
                                                 const float4* __restrict__ nodeinfo,
                                                 float* __restrict__ out) {
    float4 nu = nodeinfo[s];                    // cached gathers, L2-resident
    float4 nv = nodeinfo[r];
    float rmax = nu.z + nv.z;
    // envelope cutoff gates the whole contribution to exactly 0:
    // skip 4 transcendentals AND the atomic for ~40% of edges.
    if (xv < rmax) {
        float t = xv * (nu.y + nv.y) * INV_A_PREF;      // r/a
        float phi = 0.1818f  * __expf(-3.2f    * t)
                  + 0.5099f  * __expf(-0.9423f * t)
                  + 0.2802f  * __expf(-0.4028f * t)
                  + 0.02817f * __expf(-0.2016f * t);

        float inv_x = __frcp_rn(xv);
        float v     = KE_CONST * nu.x * nv.x * phi * inv_x;

        float rr = xv * __frcp_rn(rmax);
        float r2 = rr * rr;
        float r6 = r2 * r2 * r2;
        // 1 - 28 r^6 + 48 r^7 - 21 r^8  (p = 6), Horner
        float env = fmaf(r6, fmaf(rr, fmaf(rr, -21.0f, 48.0f), -28.0f), 1.0f);

        atomicAdd(out + r, 0.5f * v * env);     // global_atomic_add_f32
    }
}

// ------------- main edge kernel: async-to-LDS double-buffered --------------
__global__ void zbl_edge_async(const float*  __restrict__ x,
                               const int*    __restrict__ edge_index,
                               const float4* __restrict__ nodeinfo,
                               float*        __restrict__ out,
                               int n_edges, int n_tiles) {
    __shared__ alignas(16) int   s_s[2][TILE_EDGES];
    __shared__ alignas(16) int   s_r[2][TILE_EDGES];
    __shared__ alignas(16) float s_x[2][TILE_EDGES];

    const int t = threadIdx.x;                  // 0..255
    const int lane4 = t * EDGES_PER_T;          // this thread's 4 edges in tile

    // Wave-relative LDS byte offsets for this thread's 16B slices (low 32 bits
    // of the LDS-aperture generic address == logical LDS address).
    const unsigned offS[2] = { (unsigned)(uintptr_t)&s_s[0][lane4],
                               (unsigned)(uintptr_t)&s_s[1][lane4] };
    const unsigned offR[2] = { (unsigned)(uintptr_t)&s_r[0][lane4],
                               (unsigned)(uintptr_t)&s_r[1][lane4] };
    const unsigned offX[2] = { (unsigned)(uintptr_t)&s_x[0][lane4],
                               (unsigned)(uintptr_t)&s_x[1][lane4] };

    int tile = blockIdx.x;
    if (tile >= n_tiles) return;

    // Issue async stage of a tile's 3 streams (16B per thread per stream).
    auto stage = [&](int buf, int tl) {
        size_t base = (size_t)tl * TILE_EDGES + lane4;
        const int*   gs = edge_index + base;                    // senders
        const int*   gr = edge_index + (size_t)n_edges + base;  // receivers
        const float* gx = x + base;                             // distances
        asm volatile("global_load_async_to_lds_b128 %0, %1, off"
                     :: "v"(offS[buf]), "v"(gs) : "memory");
        asm volatile("global_load_async_to_lds_b128 %0, %1, off"
                     :: "v"(offR[buf]), "v"(gr) : "memory");
        asm volatile("global_load_async_to_lds_b128 %0, %1, off"
                     :: "v"(offX[buf]), "v"(gx) : "memory");
    };

    int buf = 0;
    stage(0, tile);                              // prime the pipeline

    while (true) {
        int nextTile = tile + (int)gridDim.x;
        if (nextTile < n_tiles) {
            stage(buf ^ 1, nextTile);
            // ASYNC ops complete in order: <=3 outstanding => oldest stage done.
            asm volatile("s_wait_asynccnt 0x3" ::: "memory");
        } else {
            asm volatile("s_wait_asynccnt 0x0" ::: "memory");
        }

        // Consume own 16B slices (no cross-thread data -> no barrier needed).
        int4   sv = *(const int4*)  &s_s[buf][lane4];
        int4   rv = *(const int4*)  &s_r[buf][lane4];
        float4 xv = *(const float4*)&s_x[buf][lane4];

        zbl_edge_compute(xv.x, sv.x, rv.x, nodeinfo, out);
        zbl_edge_compute(xv.y, sv.y, rv.y, nodeinfo, out);
        zbl_edge_compute(xv.z, sv.z, rv.z, nodeinfo, out);
        zbl_edge_compute(xv.w, sv.w, rv.w, nodeinfo, out);

        if (nextTile >= n_tiles) break;
        tile = nextTile;
        buf ^= 1;
    }
}

// ------------------------- tail (remainder) edges --------------------------
__global__ void zbl_edge_tail(const float*  __restrict__ x,
                              const int*    __restrict__ edge_index,
                              const float4* __restrict__ nodeinfo,
                              float*        __restrict__ out,
                              int start, int n_edges) {
    int e = start + blockIdx.x * blockDim.x + threadIdx.x;
    if (e >= n_edges) return;
    int   s  = __builtin_nontemporal_load(edge_index + e);
    int   r  = __builtin_nontemporal_load(edge_index + (size_t)n_edges + e);
    float xv = __builtin_nontemporal_load(x + e);
    zbl_edge_compute(xv, s, r, nodeinfo, out);
}

// ------------------------------- launcher ----------------------------------
extern "C" void kernel_launch(void* const* d_in, const int* in_sizes, int n_in,
                              void* d_out, int out_size, void* d_ws, size_t ws_size,
                              hipStream_t stream) {
    const float* x              = (const float*)d_in[0];
    const float* node_attrs     = (const float*)d_in[1];
    const int*   edge_index     = (const int*)d_in[2];
    const int*   atomic_numbers = (const int*)d_in[3];
    const float* covalent_radii = (const float*)d_in[4];

    const int n_edges = in_sizes[0];           // x is [E,1] -> flat E
    const int n_nodes = out_size;              // output [N]
    const int n_elem  = in_sizes[3];           // atomic_numbers [n_elem]

    float4* nodeinfo = (float4*)d_ws;          // 1.6MB scratch
    float*  out      = (float*)d_out;

    zbl_node_prep<<<(n_nodes + BLOCK_T - 1) / BLOCK_T, BLOCK_T, 0, stream>>>(
        node_attrs, atomic_numbers, covalent_radii, nodeinfo, out, n_nodes, n_elem);

    const int n_tiles = n_edges / TILE_EDGES;
    const int rem     = n_edges - n_tiles * TILE_EDGES;

    if (n_tiles > 0) {
        int grid = n_tiles < 2048 ? n_tiles : 2048;   // ~3 tiles/block pipeline
        zbl_edge_async<<<grid, BLOCK_T, 0, stream>>>(
            x, edge_index, nodeinfo, out, n_edges, n_tiles);
    }
    if (rem > 0) {
        zbl_edge_tail<<<(rem + BLOCK_T - 1) / BLOCK_T, BLOCK_T, 0, stream>>>(
            x, edge_index, nodeinfo, out, n_tiles * TILE_EDGES, n_edges);
    }
}